// Block_2645699854377
// MI455X (gfx1250) — compile-verified
//
#include <hip/hip_runtime.h>
#include <hip/hip_bf16.h>
#include <math.h>

// ---------------------------------------------------------------------------
// VMamba 3D block forward for MI455X (gfx1250, wave32).
// All GEMMs use V_WMMA_F32_16X16X4_F32 (fp32 exact path).
// B tile stored transposed in LDS so both A and B fragments are contiguous
// 8B-aligned pairs -> single ds_load_2addr_b64 per fragment, no repacking movs.
// ---------------------------------------------------------------------------

typedef float v2f __attribute__((ext_vector_type(2)));
typedef float v8f __attribute__((ext_vector_type(8)));

#define DIM      192
#define D_INNER  384
#define DM       768
#define DT_RANK  24
#define NSTATE   16
#define LSEQ     8192          // 8*32*32
#define T1       16384         // 2 * LSEQ  (tokens)
#define T2       32768         // 4 * LSEQ  (doubled batch tokens)

// ---------------------------------------------------------------------------
// WMMA fp32 GEMM:  C[M,N] = A[M,K] @ B[K,N]  (row-major, leading dims lda/ldb/ldc)
// Block: 128 threads = 4 waves; tile 64(M) x 64(N), K chunked by 16.
// Wave w computes rows [w*16, w*16+16) across all 64 columns (4 x v8f acc).
// EPI: 0=none, 1=bias+GELU(exact), 2=+residual, 3=bias+residual
// ---------------------------------------------------------------------------
#define BM 64
#define BN 64
#define BK 16
#define SA 18   // LDS row stride for A tile (conflict-free, keeps b64 align)
#define SBT 18  // LDS row stride for transposed B tile (row = N, col = K)

template <int EPI>
__global__ __launch_bounds__(128)
void gemm_wmma_f32(const float* __restrict__ A, const float* __restrict__ B,
                   float* __restrict__ C, const float* __restrict__ bias,
                   const float* __restrict__ residual,
                   int M, int N, int K, int lda, int ldb, int ldc)
{
    __shared__ float As [BM * SA];   // As[row][k]
    __shared__ float BsT[BN * SBT];  // BsT[col][k]  (transposed B tile)

    const int tid  = threadIdx.x;
    const int wave = tid >> 5;
    const int lane = tid & 31;
    const int l16  = lane & 15;
    const int koff = (lane >> 4) << 1;   // 0 for lanes 0-15, 2 for lanes 16-31
    const int bm = blockIdx.y * BM;
    const int bn = blockIdx.x * BN;

    v8f acc[4] = {};

    for (int kc = 0; kc < K; kc += BK) {
        // ---- stage A tile (64 x 16): 128 threads x 2 float4 ----
        #pragma unroll
        for (int i = 0; i < 2; ++i) {
            int idx = tid * 2 + i;           // 0..255
            int row = idx >> 2;              // 0..63
            int kq  = (idx & 3) << 2;        // 0,4,8,12
            int gk  = kc + kq;
            const float* src = A + (size_t)(bm + row) * lda + gk;
            float4 v = make_float4(0.f, 0.f, 0.f, 0.f);
            if (gk + 3 < K) {
                v = *(const float4*)src;
            } else {
                if (gk + 0 < K) v.x = src[0];
                if (gk + 1 < K) v.y = src[1];
                if (gk + 2 < K) v.z = src[2];
                if (gk + 3 < K) v.w = src[3];
            }
            float* dst = &As[row * SA + kq];
            dst[0] = v.x; dst[1] = v.y; dst[2] = v.z; dst[3] = v.w;
        }
        // ---- stage B tile (16 x 64), stored transposed: 128 threads x 2 float4 ----
        #pragma unroll
        for (int i = 0; i < 2; ++i) {
            int idx  = tid * 2 + i;          // 0..255
            int krow = idx >> 4;             // 0..15 (K within chunk)
            int nq   = (idx & 15) << 2;      // 0..60 (N within tile)
            int gk   = kc + krow;
            int gn   = bn + nq;
            float4 v = make_float4(0.f, 0.f, 0.f, 0.f);
            if (gk < K) {
                const float* src = B + (size_t)gk * ldb + gn;
                if (gn + 3 < N) {
                    v = *(const float4*)src;
                } else {
                    if (gn + 0 < N) v.x = src[0];
                    if (gn + 1 < N) v.y = src[1];
                    if (gn + 2 < N) v.z = src[2];
                    if (gn + 3 < N) v.w = src[3];
                }
            }
            BsT[(nq + 0) * SBT + krow] = v.x;
            BsT[(nq + 1) * SBT + krow] = v.y;
            BsT[(nq + 2) * SBT + krow] = v.z;
            BsT[(nq + 3) * SBT + krow] = v.w;
        }
        __syncthreads();

        // ---- compute: 4 k-steps of 4, 4 N-subtiles -> 16 WMMAs per chunk ----
        #pragma unroll
        for (int kk = 0; kk < BK; kk += 4) {
            // A frag (16x4 f32): lane<16 holds (M=l16, K=kk+0/1),
            // lane>=16 holds (M=l16, K=kk+2/3) -> contiguous b64 in LDS
            const float* ap = &As[(wave * 16 + l16) * SA + kk + koff];
            v2f a;
            a.x = ap[0];
            a.y = ap[1];
            #pragma unroll
            for (int j = 0; j < 4; ++j) {
                // B frag (4x16 f32): lane<16 holds (N=l16, K=kk+0/1), etc.
                // Transposed layout -> contiguous 8B-aligned pair
                const float* bp = &BsT[(j * 16 + l16) * SBT + kk + koff];
                v2f b;
                b.x = bp[0];
                b.y = bp[1];
                acc[j] = __builtin_amdgcn_wmma_f32_16x16x4_f32(
                    false, a, false, b, (short)0, acc[j], false, false);
            }
        }
        __syncthreads();
    }

    // ---- epilogue: C/D layout: vgpr r -> (M = r + 8*(lane>=16), N = l16) ----
    const int rbase = bm + wave * 16 + ((lane >> 4) << 3);
    #pragma unroll
    for (int j = 0; j < 4; ++j) {
        int col = bn + j * 16 + l16;
        if (col >= N) continue;
        #pragma unroll
        for (int r = 0; r < 8; ++r) {
            int row = rbase + r;
            float v = acc[j][r];
            if (EPI == 1) {               // bias + exact GELU
                v += bias[col];
                v = 0.5f * v * (1.f + erff(v * 0.70710678118654752f));
            } else if (EPI == 2) {        // + residual
                v += residual[(size_t)row * ldc + col];
            } else if (EPI == 3) {        // bias + residual
                v += bias[col] + residual[(size_t)row * ldc + col];
            }
            C[(size_t)row * ldc + col] = v;
        }
    }
}

// ---------------------------------------------------------------------------
// LayerNorm over last dim (192). One wave per row, 8 rows per 256-thr block.
// ---------------------------------------------------------------------------
__global__ __launch_bounds__(256)
void ln_kernel(const float* __restrict__ x, const float* __restrict__ w,
               const float* __restrict__ b, float* __restrict__ out, int M)
{
    int wave = threadIdx.x >> 5;
    int lane = threadIdx.x & 31;
    int row  = blockIdx.x * 8 + wave;
    if (row >= M) return;
    const float* xr = x + (size_t)row * DIM;
    float v[6];
    float s = 0.f;
    #pragma unroll
    for (int i = 0; i < 6; ++i) { v[i] = xr[lane + 32 * i]; s += v[i]; }
    #pragma unroll
    for (int o = 16; o > 0; o >>= 1) s += __shfl_xor(s, o);
    float mean = s * (1.f / DIM);
    float vs = 0.f;
    #pragma unroll
    for (int i = 0; i < 6; ++i) { float d = v[i] - mean; vs += d * d; }
    #pragma unroll
    for (int o = 16; o > 0; o >>= 1) vs += __shfl_xor(vs, o);
    float inv = rsqrtf(vs * (1.f / DIM) + 1e-5f);
    float* orow = out + (size_t)row * DIM;
    #pragma unroll
    for (int i = 0; i < 6; ++i) {
        int c = lane + 32 * i;
        orow[c] = (v[i] - mean) * inv * w[c] + b[c];
    }
}

// ---------------------------------------------------------------------------
// Depthwise 3x3x3 conv (pad 1) + bias + SiLU on xs = xz[:, 0:384].
// Writes both the forward sequence and the reversed copy (seq2).
// ---------------------------------------------------------------------------
__global__ __launch_bounds__(256)
void conv3d_silu_kernel(const float* __restrict__ xz, const float* __restrict__ w,
                        const float* __restrict__ bias, float* __restrict__ seq2)
{
    int gid = blockIdx.x * 256 + threadIdx.x;   // T1 * 384 threads
    int c   = gid % D_INNER;
    int tok = gid / D_INNER;
    int b   = tok >> 13;
    int rem = tok & (LSEQ - 1);
    int dz  = rem >> 10;
    int hy  = (rem >> 5) & 31;
    int wx  = rem & 31;
    float acc = bias[c];
    const float* wc = w + c * 27;
    #pragma unroll
    for (int kd = 0; kd < 3; ++kd) {
        int zz = dz + kd - 1;
        if (zz < 0 || zz >= 8) continue;
        #pragma unroll
        for (int kh = 0; kh < 3; ++kh) {
            int yy = hy + kh - 1;
            if (yy < 0 || yy >= 32) continue;
            #pragma unroll
            for (int kw = 0; kw < 3; ++kw) {
                int xx = wx + kw - 1;
                if (xx < 0 || xx >= 32) continue;
                int t2 = ((b * 8 + zz) * 32 + yy) * 32 + xx;
                acc += wc[(kd * 3 + kh) * 3 + kw] * xz[(size_t)t2 * (2 * D_INNER) + c];
            }
        }
    }
    acc = acc / (1.f + expf(-acc));   // SiLU
    int l = rem;
    seq2[((size_t)(b * LSEQ) + l) * D_INNER + c] = acc;
    seq2[((size_t)((b + 2) * LSEQ) + (LSEQ - 1 - l)) * D_INNER + c] = acc;
}

// ---------------------------------------------------------------------------
// Causal depthwise conv1d (k=4, left pad 3) + bias + SiLU.  xm -> u
// ---------------------------------------------------------------------------
__global__ __launch_bounds__(256)
void conv1d_silu_kernel(const float* __restrict__ xm, const float* __restrict__ w,
                        const float* __restrict__ bias, float* __restrict__ u)
{
    int gid = blockIdx.x * 256 + threadIdx.x;   // T2 * 768 threads
    int c   = gid % DM;
    int tok = gid / DM;                         // b2*L + l
    int l   = tok & (LSEQ - 1);
    float acc = bias[c];
    const float* wc = w + c * 4;
    #pragma unroll
    for (int j = 0; j < 4; ++j) {
        int li = l + j - 3;
        if (li >= 0) acc += wc[j] * xm[(size_t)(tok + j - 3) * DM + c];
    }
    acc = acc / (1.f + expf(-acc));
    u[(size_t)gid] = acc;
}

// ---------------------------------------------------------------------------
// Selective scan: one thread per (batch2, channel).  16 states in registers.
// Epilogue: y = (scan + u*Dp) * SiLU(zm), written in-place over zm.
// ---------------------------------------------------------------------------
__global__ __launch_bounds__(256)
void scan_kernel(const float* __restrict__ dt, const float* __restrict__ dt_bias,
                 const float* __restrict__ A_log, const float* __restrict__ u,
                 const float* __restrict__ xdbl, const float* __restrict__ Dp,
                 float* __restrict__ zm_ym)
{
    int tid = blockIdx.x * 256 + threadIdx.x;   // 0..3071
    int b   = tid / DM;
    int d   = tid % DM;

    float Aa[NSTATE];
    #pragma unroll
    for (int n = 0; n < NSTATE; ++n) Aa[n] = -expf(A_log[d * NSTATE + n]);
    float st[NSTATE];
    #pragma unroll
    for (int n = 0; n < NSTATE; ++n) st[n] = 0.f;

    const float dbias = dt_bias[d];
    const float Dd    = Dp[d];
    const size_t base = (size_t)b * LSEQ;

    for (int l = 0; l < LSEQ; ++l) {
        size_t idx = base + l;
        float dtv   = dt[idx * DM + d] + dbias;
        float delta = (dtv > 20.f) ? dtv : log1pf(expf(dtv));   // softplus
        float uv    = u[idx * DM + d];
        float du    = delta * uv;
        const float* bc = xdbl + idx * (DT_RANK + 2 * NSTATE);
        float y = 0.f;
        #pragma unroll
        for (int n = 0; n < NSTATE; ++n) {
            float dA = expf(delta * Aa[n]);
            st[n] = dA * st[n] + du * bc[DT_RANK + n];
            y += st[n] * bc[DT_RANK + NSTATE + n];
        }
        float z  = zm_ym[idx * DM + d];
        float sz = z / (1.f + expf(-z));
        zm_ym[idx * DM + d] = (y + uv * Dd) * sz;
    }
}

// ---------------------------------------------------------------------------
// Combine forward + reversed scan outputs, multiply by SiLU(z from xz[:,384:]).
// ---------------------------------------------------------------------------
__global__ __launch_bounds__(256)
void combine_kernel(const float* __restrict__ mout, const float* __restrict__ xz,
                    float* __restrict__ ssd)
{
    int gid = blockIdx.x * 256 + threadIdx.x;   // T1 * 384
    int c = gid % D_INNER;
    int t = gid / D_INNER;
    int b = t >> 13;
    int l = t & (LSEQ - 1);
    float yf = mout[((size_t)(b * LSEQ) + l) * D_INNER + c];
    float yr = mout[((size_t)((b + 2) * LSEQ) + (LSEQ - 1 - l)) * D_INNER + c];
    float z  = xz[(size_t)t * (2 * D_INNER) + D_INNER + c];
    float sz = z / (1.f + expf(-z));
    ssd[(size_t)gid] = (yf + yr) * sz;
}

// ---------------------------------------------------------------------------
// Host launcher
// ---------------------------------------------------------------------------
extern "C" void kernel_launch(void* const* d_in, const int* in_sizes, int n_in,
                              void* d_out, int out_size, void* d_ws, size_t ws_size,
                              hipStream_t stream)
{
    const float* x_in      = (const float*)d_in[0];
    const float* norm1_w   = (const float*)d_in[1];
    const float* norm1_b   = (const float*)d_in[2];
    const float* in_proj_w = (const float*)d_in[3];
    const float* conv3_w   = (const float*)d_in[4];
    const float* conv3_b   = (const float*)d_in[5];
    const float* m_in_w    = (const float*)d_in[6];
    const float* m_conv_w  = (const float*)d_in[7];
    const float* m_conv_b  = (const float*)d_in[8];
    const float* x_proj_w  = (const float*)d_in[9];
    const float* dt_proj_w = (const float*)d_in[10];
    const float* dt_proj_b = (const float*)d_in[11];
    const float* A_log     = (const float*)d_in[12];
    const float* Dp        = (const float*)d_in[13];
    const float* m_out_w   = (const float*)d_in[14];
    const float* out_proj_w= (const float*)d_in[15];
    const float* norm2_w   = (const float*)d_in[16];
    const float* norm2_b   = (const float*)d_in[17];
    const float* fc1_w     = (const float*)d_in[18];
    const float* fc1_b     = (const float*)d_in[19];
    const float* fc2_w     = (const float*)d_in[20];
    const float* fc2_b     = (const float*)d_in[21];
    float* out = (float*)d_out;

    // Workspace layout (floats); buffers reused once dead.
    float* ws = (float*)d_ws;
    size_t o = 0;
    float* f_H    = ws + o; o += (size_t)T1 * DIM;        // h, then h2
    float* f_XZ   = ws + o; o += (size_t)T1 * (2*D_INNER);// xz, then a1 (fc1 out)
    float* f_SEQ2 = ws + o; o += (size_t)T2 * D_INNER;    // seq2, then m_out result
    float* f_XM   = ws + o; o += (size_t)T2 * DM;         // xm, then dt
    float* f_ZM   = ws + o; o += (size_t)T2 * DM;         // zm, then ym (in place)
    float* f_U    = ws + o; o += (size_t)T2 * DM;         // conv1d out u, then ssd
    float* f_XDBL = ws + o; o += (size_t)T2 * (DT_RANK + 2*NSTATE);
    float* f_X1   = ws + o; o += (size_t)T1 * DIM;        // residual after ssm
    (void)o; (void)ws_size; (void)in_sizes; (void)n_in; (void)out_size;

    const dim3 blk128(128), blk256(256);

    // 1. LN1: x -> h
    ln_kernel<<<T1 / 8, blk256, 0, stream>>>(x_in, norm1_w, norm1_b, f_H, T1);

    // 2. in_proj: h[T1,192] @ [192,768] -> xz
    gemm_wmma_f32<0><<<dim3((2*D_INNER)/BN, T1/BM), blk128, 0, stream>>>(
        f_H, in_proj_w, f_XZ, nullptr, nullptr, T1, 2*D_INNER, DIM, DIM, 2*D_INNER, 2*D_INNER);

    // 3. depthwise conv3d + SiLU -> seq2 (forward + reversed)
    conv3d_silu_kernel<<<(size_t)T1 * D_INNER / 256, blk256, 0, stream>>>(
        f_XZ, conv3_w, conv3_b, f_SEQ2);

    // 4/5. m_in: seq2[T2,384] @ m_in_w -> xm (cols 0..767), zm (cols 768..1535)
    gemm_wmma_f32<0><<<dim3(DM/BN, T2/BM), blk128, 0, stream>>>(
        f_SEQ2, m_in_w, f_XM, nullptr, nullptr, T2, DM, D_INNER, D_INNER, 2*DM, DM);
    gemm_wmma_f32<0><<<dim3(DM/BN, T2/BM), blk128, 0, stream>>>(
        f_SEQ2, m_in_w + DM, f_ZM, nullptr, nullptr, T2, DM, D_INNER, D_INNER, 2*DM, DM);

    // 6. causal depthwise conv1d + SiLU: xm -> u
    conv1d_silu_kernel<<<(size_t)T2 * DM / 256, blk256, 0, stream>>>(
        f_XM, m_conv_w, m_conv_b, f_U);

    // 7. x_proj: u[T2,768] @ [768,56] -> x_dbl
    gemm_wmma_f32<0><<<dim3(1, T2/BM), blk128, 0, stream>>>(
        f_U, x_proj_w, f_XDBL, nullptr, nullptr,
        T2, DT_RANK + 2*NSTATE, DM, DM, DT_RANK + 2*NSTATE, DT_RANK + 2*NSTATE);

    // 8. dt_proj: x_dbl[:, 0:24] @ [24,768] -> dt (reuse f_XM)
    gemm_wmma_f32<0><<<dim3(DM/BN, T2/BM), blk128, 0, stream>>>(
        f_XDBL, dt_proj_w, f_XM, nullptr, nullptr,
        T2, DM, DT_RANK, DT_RANK + 2*NSTATE, DM, DM);

    // 9. selective scan (ym written in place over zm)
    scan_kernel<<<(4 * DM) / 256, blk256, 0, stream>>>(
        f_XM, dt_proj_b, A_log, f_U, f_XDBL, Dp, f_ZM);

    // 10. m_out: ym[T2,768] @ [768,384] -> (reuse f_SEQ2)
    gemm_wmma_f32<0><<<dim3(D_INNER/BN, T2/BM), blk128, 0, stream>>>(
        f_ZM, m_out_w, f_SEQ2, nullptr, nullptr, T2, D_INNER, DM, DM, D_INNER, D_INNER);

    // 11. combine fwd+rev, * SiLU(z) -> ssd (reuse f_U)
    combine_kernel<<<(size_t)T1 * D_INNER / 256, blk256, 0, stream>>>(
        f_SEQ2, f_XZ, f_U);

    // 12. out_proj + residual(x): ssd[T1,384] @ [384,192] + x -> x1
    gemm_wmma_f32<2><<<dim3(DIM/BN, T1/BM), blk128, 0, stream>>>(
        f_U, out_proj_w, f_X1, nullptr, x_in, T1, DIM, D_INNER, D_INNER, DIM, DIM);

    // 13. LN2: x1 -> h2 (reuse f_H)
    ln_kernel<<<T1 / 8, blk256, 0, stream>>>(f_X1, norm2_w, norm2_b, f_H, T1);

    // 14. fc1 + bias + GELU: h2[T1,192] @ [192,768] -> a1 (reuse f_XZ)
    gemm_wmma_f32<1><<<dim3((4*DIM)/BN, T1/BM), blk128, 0, stream>>>(
        f_H, fc1_w, f_XZ, fc1_b, nullptr, T1, 4*DIM, DIM, DIM, 4*DIM, 4*DIM);

    // 15. fc2 + bias + residual(x1): a1[T1,768] @ [768,192] -> out
    gemm_wmma_f32<3><<<dim3(DIM/BN, T1/BM), blk128, 0, stream>>>(
        f_XZ, fc2_w, out, fc2_b, f_X1, T1, DIM, 4*DIM, 4*DIM, DIM, DIM);
}